// Attention_15917148799490
// MI455X (gfx1250) — compile-verified
//
#include <hip/hip_runtime.h>
#include <math.h>

// ---------------------------------------------------------------------------
// Problem constants
// ---------------------------------------------------------------------------
#define BB   16
#define HH   56
#define WW   56
#define CC   384
#define C2   192
#define NHD  8
#define WIN  7
#define HD2  24
#define NTOK (BB * HH * WW)          // 50176 tokens
#define NPOOL (BB * WIN * WIN)       // 784 pooled tokens
#define ATT_SCALE 0.20412414523193154f  // 24^-0.5

typedef float v2f __attribute__((ext_vector_type(2)));
typedef float v8f __attribute__((ext_vector_type(8)));

enum { EPI_NONE = 0, EPI_GELU = 1, EPI_MUL = 2 };

// ---------------------------------------------------------------------------
// LayerNorm: one wave32 per token
// ---------------------------------------------------------------------------
__global__ __launch_bounds__(256) void ln_kernel(
    const float* __restrict__ x, const float* __restrict__ w,
    const float* __restrict__ b, float* __restrict__ out, int C, int ntok)
{
    const int lane = threadIdx.x & 31;
    const int wv   = threadIdx.x >> 5;
    const int tok  = blockIdx.x * 8 + wv;
    if (tok >= ntok) return;
    const float* xr = x + (size_t)tok * C;
    float s = 0.f, ss = 0.f;
    for (int c = lane; c < C; c += 32) { float v = xr[c]; s += v; ss += v * v; }
    #pragma unroll
    for (int off = 16; off; off >>= 1) {
        s  += __shfl_xor(s,  off, 32);
        ss += __shfl_xor(ss, off, 32);
    }
    const float mean = s / C;
    const float var  = ss / C - mean * mean;
    const float rstd = rsqrtf(var + 1e-6f);
    float* orow = out + (size_t)tok * C;
    for (int c = lane; c < C; c += 32)
        orow[c] = (xr[c] - mean) * rstd * w[c] + b[c];
}

// ---------------------------------------------------------------------------
// 8x8 average pool of concat(xn, xen) -> pooled [784, 576]
// ---------------------------------------------------------------------------
__global__ __launch_bounds__(576) void pool_kernel(
    const float* __restrict__ xn, const float* __restrict__ xen,
    float* __restrict__ pooled)
{
    const int blk = blockIdx.x;            // 0..783
    const int b  = blk / 49;
    const int q  = blk % 49;
    const int py = q / 7, px = q % 7;
    const int c  = threadIdx.x;            // 0..575
    float s = 0.f;
    #pragma unroll
    for (int i = 0; i < 8; ++i) {
        const int y = py * 8 + i;
        #pragma unroll
        for (int j = 0; j < 8; ++j) {
            const int x = px * 8 + j;
            const size_t tok = (size_t)((b * HH + y) * WW + x);
            s += (c < CC) ? xn[tok * CC + c] : xen[tok * C2 + (c - CC)];
        }
    }
    pooled[(size_t)blk * (CC + C2) + c] = s * (1.0f / 64.0f);
}

// ---------------------------------------------------------------------------
// fp32 WMMA GEMM with async global->LDS double buffering.
//   out[m, outOff+n] = epi(A[m,:K] @ W[:K, n] + bias[n])
// Block: 256 threads (8 waves). Block tile 128x64, K chunked by 64.
// LDS tiles staged with global_load_async_to_lds_b128 (ASYNCcnt), ping-pong.
//   A fragment (16x4 f32): lane = (k>=2)*16 + m , vgpr = k&1  (ds_load_b64)
//   B fragment (4x16 f32): lane = (k>=2)*16 + n , vgpr = k&1  (ds_load_b32 x2)
//   C/D (16x16 f32): element (M = r + (lane>=16)*8, N = lane&15) in vgpr r
// Requires: M % 16 == 0, K % 64 == 0, N % 64 == 0.
// ---------------------------------------------------------------------------
#define KC  64
#define SAS 68   // sA row stride (floats): 16B-aligned rows, conflict-free b64 reads
#define SBS 68   // sB row stride (floats): 16B-aligned rows

__global__ __launch_bounds__(256) void gemm_f32_wmma(
    const float* __restrict__ A, const float* __restrict__ W,
    const float* __restrict__ bias, const float* __restrict__ aux,
    float* __restrict__ out, int M, int K, int N, int ldo, int outOff, int mode)
{
    __shared__ float sA[2][128 * SAS];
    __shared__ float sB[2][KC * SBS];

    const int tid   = threadIdx.x;
    const int lane  = tid & 31;
    const int wv    = tid >> 5;
    const int mBase = blockIdx.x * 128;
    const int nBase = blockIdx.y * 64;
    const int mloc  = lane & 15;
    const int kgrp  = lane >> 4;           // 0 or 1 -> k pair {0,1} or {2,3}
    const int klo   = kgrp * 2;
    const int nChunks = K / KC;

    // Issue async copies of one K-chunk into LDS buffer `buf` (12 per thread).
    auto issue = [&](int buf, int kc) {
        #pragma unroll
        for (int i = 0; i < 8; ++i) {          // A: 128 rows x 16 x 16B
            const int idx = i * 256 + tid;
            const int r = idx >> 4, c4 = idx & 15;
            int gr = mBase + r; if (gr >= M) gr = M - 1;
            const float* g = A + (size_t)gr * K + kc + c4 * 4;
            const unsigned dst = (unsigned)(uintptr_t)(&sA[buf][r * SAS + c4 * 4]);
            asm volatile("global_load_async_to_lds_b128 %0, %1, off"
                         :: "v"(dst), "v"(g) : "memory");
        }
        #pragma unroll
        for (int i = 0; i < 4; ++i) {          // B: 64 rows x 16 x 16B
            const int idx = i * 256 + tid;
            const int r = idx >> 4, c4 = idx & 15;
            const float* g = W + (size_t)(kc + r) * N + nBase + c4 * 4;
            const unsigned dst = (unsigned)(uintptr_t)(&sB[buf][r * SBS + c4 * 4]);
            asm volatile("global_load_async_to_lds_b128 %0, %1, off"
                         :: "v"(dst), "v"(g) : "memory");
        }
    };

    v8f acc0 = {}, acc1 = {}, acc2 = {}, acc3 = {};

    issue(0, 0);
    for (int ic = 0; ic < nChunks; ++ic) {
        const int buf = ic & 1;
        if (ic + 1 < nChunks) {
            issue(buf ^ 1, (ic + 1) * KC);     // prefetch next chunk
            // async completes in order: <=12 outstanding => current chunk landed
            asm volatile("s_wait_asynccnt 0xc" ::: "memory");
        } else {
            asm volatile("s_wait_asynccnt 0x0" ::: "memory");
        }
        __syncthreads();

        const float* a_ = &sA[buf][(wv * 16 + mloc) * SAS + klo];
        const float* b_ = &sB[buf][klo * SBS + mloc];
        #pragma unroll
        for (int k0 = 0; k0 < KC; k0 += 4) {
            const v2f a2 = *reinterpret_cast<const v2f*>(a_ + k0);
            const float* bp = b_ + k0 * SBS;
            v2f b2;
            b2.x = bp[0];        b2.y = bp[SBS];
            acc0 = __builtin_amdgcn_wmma_f32_16x16x4_f32(false, a2, false, b2, (short)0, acc0, false, false);
            b2.x = bp[16];       b2.y = bp[SBS + 16];
            acc1 = __builtin_amdgcn_wmma_f32_16x16x4_f32(false, a2, false, b2, (short)0, acc1, false, false);
            b2.x = bp[32];       b2.y = bp[SBS + 32];
            acc2 = __builtin_amdgcn_wmma_f32_16x16x4_f32(false, a2, false, b2, (short)0, acc2, false, false);
            b2.x = bp[48];       b2.y = bp[SBS + 48];
            acc3 = __builtin_amdgcn_wmma_f32_16x16x4_f32(false, a2, false, b2, (short)0, acc3, false, false);
        }
        __syncthreads();
    }

    const int mTile = mBase + wv * 16;
    #pragma unroll
    for (int nt = 0; nt < 4; ++nt) {
        const v8f acc = (nt == 0) ? acc0 : (nt == 1) ? acc1 : (nt == 2) ? acc2 : acc3;
        const int ncol = nBase + nt * 16 + mloc;
        const float bb = bias ? bias[ncol] : 0.f;
        #pragma unroll
        for (int r = 0; r < 8; ++r) {
            const int orow = mTile + r + kgrp * 8;
            if (orow >= M) continue;
            float v = acc[r] + bb;
            if (mode == EPI_GELU) {
                v = 0.5f * v * (1.0f + erff(v * 0.70710678118654752f));
            } else if (mode == EPI_MUL) {
                v *= aux[(size_t)orow * N + ncol];
            }
            out[(size_t)orow * ldo + outOff + ncol] = v;
        }
    }
}

// ---------------------------------------------------------------------------
// Depthwise 7x7 conv, padding 3, channels-last [B,H,W,Cch]; block = Cch threads
// ---------------------------------------------------------------------------
__global__ void dwconv_kernel(
    const float* __restrict__ in, const float* __restrict__ w,
    const float* __restrict__ bias, float* __restrict__ out, int Cch)
{
    const int blk = blockIdx.x;            // token
    const int b = blk / (HH * WW);
    const int r = blk % (HH * WW);
    const int y = r / WW, x = r % WW;
    const int c = threadIdx.x;
    float acc = bias[c];
    #pragma unroll
    for (int ky = 0; ky < 7; ++ky) {
        const int yy = y + ky - 3;
        if (yy < 0 || yy >= HH) continue;
        #pragma unroll
        for (int kx = 0; kx < 7; ++kx) {
            const int xx = x + kx - 3;
            if (xx < 0 || xx >= WW) continue;
            acc += in[((size_t)((b * HH + yy) * WW + xx)) * Cch + c] * w[c * 49 + ky * 7 + kx];
        }
    }
    out[(size_t)blk * Cch + c] = acc;
}

// ---------------------------------------------------------------------------
// Attention: one wave32 per (b, head, query); online softmax over 3136 keys
// m: [B,49, NH*24]; kv: [B*3136, 2*NH*24]; o: [B*NH*49, 24]
// ---------------------------------------------------------------------------
__global__ __launch_bounds__(256) void attn_kernel(
    const float* __restrict__ m, const float* __restrict__ kv,
    float* __restrict__ o)
{
    const int lane = threadIdx.x & 31;
    const int wv   = threadIdx.x >> 5;
    const int gw   = blockIdx.x * 8 + wv;       // 0..6271
    if (gw >= BB * NHD * 49) return;
    const int bh = gw / 49, q = gw % 49;
    const int b = bh / NHD, h = bh % NHD;

    const float* mq = m + ((size_t)(b * 49 + q)) * C2 + h * HD2;
    float qv[HD2];
    #pragma unroll
    for (int d = 0; d < HD2; ++d) qv[d] = mq[d] * ATT_SCALE;

    const float* kvb = kv + (size_t)b * (HH * WW) * CC + h * HD2;
    float mmax = -INFINITY, l = 0.f;
    float ov[HD2];
    #pragma unroll
    for (int d = 0; d < HD2; ++d) ov[d] = 0.f;

    for (int t = lane; t < HH * WW; t += 32) {
        const float* kr = kvb + (size_t)t * CC;
        float s = 0.f;
        #pragma unroll
        for (int d = 0; d < HD2; ++d) s += qv[d] * kr[d];
        const float mnew  = fmaxf(mmax, s);
        const float alpha = __expf(mmax - mnew);
        const float p     = __expf(s - mnew);
        l = l * alpha + p;
        const float* vr = kr + C2;
        #pragma unroll
        for (int d = 0; d < HD2; ++d) ov[d] = ov[d] * alpha + p * vr[d];
        mmax = mnew;
    }

    #pragma unroll
    for (int off = 16; off; off >>= 1) {
        const float m2 = __shfl_xor(mmax, off, 32);
        const float l2 = __shfl_xor(l,    off, 32);
        const float mn = fmaxf(mmax, m2);
        const float a1 = __expf(mmax - mn);
        const float a2 = __expf(m2   - mn);
        l = l * a1 + l2 * a2;
        #pragma unroll
        for (int d = 0; d < HD2; ++d) {
            const float o2 = __shfl_xor(ov[d], off, 32);
            ov[d] = ov[d] * a1 + o2 * a2;
        }
        mmax = mn;
    }
    const float inv = 1.0f / l;
    float* orow = o + ((size_t)(bh * 49 + q)) * HD2;
    #pragma unroll
    for (int d = 0; d < HD2; ++d)
        if (lane == d) orow[d] = ov[d] * inv;
}

// ---------------------------------------------------------------------------
// Bilinear 7->56 upsample (half-pixel centers, edge clamp)
// o: [B*NH*49, 24] ; writes cat[:, 384:576]
// ---------------------------------------------------------------------------
__global__ __launch_bounds__(C2) void upsample_kernel(
    const float* __restrict__ o, float* __restrict__ cat)
{
    const int tok = blockIdx.x;
    const int b = tok / (HH * WW);
    const int r = tok % (HH * WW);
    const int y = r / WW, x = r % WW;
    const int c = threadIdx.x;               // 0..191
    const int h = c / HD2, d = c % HD2;

    const float fy = (y + 0.5f) * 0.125f - 0.5f;
    const float fx = (x + 0.5f) * 0.125f - 0.5f;
    const int y0 = (int)floorf(fy);
    const int x0 = (int)floorf(fx);
    const float wy = fy - (float)y0;
    const float wx = fx - (float)x0;
    const int y0c = min(max(y0, 0), 6), y1c = min(max(y0 + 1, 0), 6);
    const int x0c = min(max(x0, 0), 6), x1c = min(max(x0 + 1, 0), 6);

    const float* ob = o + ((size_t)(b * NHD + h) * 49) * HD2 + d;
    const float v00 = ob[(y0c * 7 + x0c) * HD2];
    const float v01 = ob[(y0c * 7 + x1c) * HD2];
    const float v10 = ob[(y1c * 7 + x0c) * HD2];
    const float v11 = ob[(y1c * 7 + x1c) * HD2];
    const float v = (1.f - wy) * ((1.f - wx) * v00 + wx * v01)
                  +        wy  * ((1.f - wx) * v10 + wx * v11);
    cat[(size_t)tok * (2 * CC) + CC + c] = v;
}

// ---------------------------------------------------------------------------
// Host-side launcher
// ---------------------------------------------------------------------------
static void launch_gemm(const float* A, const float* W, const float* bias,
                        const float* aux, float* out, int M, int K, int N,
                        int ldo, int outOff, int mode, hipStream_t s)
{
    dim3 grid((M + 127) / 128, N / 64);
    gemm_f32_wmma<<<grid, 256, 0, s>>>(A, W, bias, aux, out, M, K, N, ldo, outOff, mode);
}

extern "C" void kernel_launch(void* const* d_in, const int* in_sizes, int n_in,
                              void* d_out, int out_size, void* d_ws, size_t ws_size,
                              hipStream_t stream)
{
    const float* x       = (const float*)d_in[0];
    const float* x_e     = (const float*)d_in[1];
    const float* norm_w  = (const float*)d_in[2];
    const float* norm_b  = (const float*)d_in[3];
    const float* norme_w = (const float*)d_in[4];
    const float* norme_b = (const float*)d_in[5];
    const float* q_w     = (const float*)d_in[6];
    const float* q_b     = (const float*)d_in[7];
    const float* qcut_w  = (const float*)d_in[8];
    const float* qcut_b  = (const float*)d_in[9];
    const float* a_w     = (const float*)d_in[10];
    const float* a_b     = (const float*)d_in[11];
    const float* l_w     = (const float*)d_in[12];
    const float* l_b     = (const float*)d_in[13];
    const float* conv_w  = (const float*)d_in[14];
    const float* conv_b  = (const float*)d_in[15];
    const float* econv_w = (const float*)d_in[16];
    const float* econv_b = (const float*)d_in[17];
    const float* efore_w = (const float*)d_in[18];
    const float* efore_b = (const float*)d_in[19];
    const float* eback_w = (const float*)d_in[20];
    const float* eback_b = (const float*)d_in[21];
    const float* kv_w    = (const float*)d_in[22];
    const float* kv_b    = (const float*)d_in[23];
    const float* scl_w   = (const float*)d_in[24];
    const float* scl_b   = (const float*)d_in[25];
    const float* proj_w  = (const float*)d_in[26];
    const float* proj_b  = (const float*)d_in[27];
    const float* proje_w = (const float*)d_in[28];
    const float* proje_b = (const float*)d_in[29];
    (void)in_sizes; (void)n_in; (void)out_size; (void)ws_size;

    // Workspace layout (floats)
    float* ws     = (float*)d_ws;
    const size_t NT = NTOK;
    float* xn     = ws;                     // [NT,384] layernormed x
    float* xen    = xn     + NT * CC;       // [NT,192] layernormed x_e
    float* qb     = xen    + NT * C2;       // [NT,384] q
    float* cutb   = qb     + NT * CC;       // [NT,192] cut (pre-multiply)
    float* lxb    = cutb   + NT * C2;       // [NT,384] gelu(xn@l_w)
    float* efb    = lxb    + NT * CC;       // [NT,192] ef
    float* clx    = efb    + NT * C2;       // [NT,384] dwconv(lx)
    float* efc    = clx    + NT * CC;       // [NT,192] dwconv(ef)
    float* kvb    = efc    + NT * C2;       // [NT,384] kv
    float* cat    = kvb    + NT * CC;       // [NT,768] [g | attn | cut*xe2]
    float* pooled = cat    + NT * 2 * CC;   // [784,576]
    float* mbuf   = pooled + (size_t)NPOOL * (CC + C2);   // [784,192]
    float* obuf   = mbuf   + (size_t)NPOOL * C2;          // [B*NH*49,24]

    float* x_out   = (float*)d_out;                 // [NT,384]
    float* x_e_out = x_out + NT * CC;               // [NT,192]

    // 1) LayerNorms (one wave per token)
    ln_kernel<<<(NTOK + 7) / 8, 256, 0, stream>>>(x,   norm_w,  norm_b,  xn,  CC, NTOK);
    ln_kernel<<<(NTOK + 7) / 8, 256, 0, stream>>>(x_e, norme_w, norme_b, xen, C2, NTOK);

    // 2) 8x8 average pool of concat(xn, xen)
    pool_kernel<<<NPOOL, CC + C2, 0, stream>>>(xn, xen, pooled);

    // 3) Token GEMMs from xn / xen
    launch_gemm(xn,  q_w,     q_b,     nullptr, qb,   NTOK, CC, CC, CC, 0, EPI_NONE, stream);
    launch_gemm(xn,  qcut_w,  qcut_b,  nullptr, cutb, NTOK, CC, C2, C2, 0, EPI_NONE, stream);
    launch_gemm(xn,  l_w,     l_b,     nullptr, lxb,  NTOK, CC, CC, CC, 0, EPI_GELU, stream);
    launch_gemm(xen, efore_w, efore_b, nullptr, efb,  NTOK, C2, C2, C2, 0, EPI_NONE, stream);

    // 4) Pooled -> shortcut queries m
    launch_gemm(pooled, scl_w, scl_b, nullptr, mbuf, NPOOL, CC + C2, C2, C2, 0, EPI_NONE, stream);

    // 5) Depthwise convs
    dwconv_kernel<<<NTOK, CC, 0, stream>>>(lxb, conv_w,  conv_b,  clx, CC);
    dwconv_kernel<<<NTOK, C2, 0, stream>>>(efb, econv_w, econv_b, efc, C2);

    // 6) GEMMs with fused epilogues into the concat buffer
    //    g = (dwconv(lx)@a_w + a_b) * q       -> cat[:, 0:384]
    launch_gemm(clx, a_w, a_b, qb, cat, NTOK, CC, CC, 2 * CC, 0, EPI_MUL, stream);
    //    kv = lx@kv_w + kv_b
    launch_gemm(lxb, kv_w, kv_b, nullptr, kvb, NTOK, CC, CC, CC, 0, EPI_NONE, stream);
    //    cut * xe2 = cut * (dwconv(ef)@eback_w + eback_b) -> cat[:, 576:768]
    launch_gemm(efc, eback_w, eback_b, cutb, cat, NTOK, C2, C2, 2 * CC, 3 * C2, EPI_MUL, stream);

    // 7) Attention (49 queries x 3136 keys per (b,head)), then bilinear 7->56
    attn_kernel<<<(BB * NHD * 49) / 8, 256, 0, stream>>>(mbuf, kvb, obuf);
    upsample_kernel<<<NTOK, C2, 0, stream>>>(obuf, cat);   // -> cat[:, 384:576]

    // 8) Output projections
    launch_gemm(cat, proj_w,  proj_b,  nullptr, x_out,   NTOK, 2 * CC, CC, CC, 0, EPI_NONE, stream);
    launch_gemm(cat, proje_w, proje_b, nullptr, x_e_out, NTOK, 2 * CC, C2, C2, 0, EPI_NONE, stream);
}